// HeatmapHead_42571715838609
// MI455X (gfx1250) — compile-verified
//
#include <hip/hip_runtime.h>
#include <hip/hip_bf16.h>
#include <math.h>

typedef float v2f __attribute__((ext_vector_type(2)));
typedef float v8f __attribute__((ext_vector_type(8)));

#define INCH 640
#define OUTCH 133
#define HH 96
#define WW 128
#define SP (HH * WW)        // 12288
#define IH 48
#define IW 64
#define EPS_F32 1.1920928955078125e-7f
#define PFDIST 8            // prefetch distance in K-channels

// ---------------------------------------------------------------------------
// Stage A: ConvTranspose2d(640->640, k=4, s=2, p=1) as 4 parity GEMMs on WMMA.
// Each wave: 16 out-ch x 64 spatial (4 N-tiles, register-blocked so the A
// (weights) fetch is reused by 4 V_WMMA_F32_16X16X4_F32 ops per channel).
// ---------------------------------------------------------------------------
__global__ void __launch_bounds__(256) hh_deconv_wmma(
    const float* __restrict__ X, const float* __restrict__ Wd,
    float* __restrict__ Out) {
  int t = blockIdx.x * 8 + (threadIdx.x >> 5);
  int i  = t % 48;       t /= 48;   // input row
  int ot = t % 40;       t /= 40;   // out-channel tile (640/16)
  int bb = t & 1;        t >>= 1;   // batch
  int par = t;                      // 0..3
  int a  = par >> 1;                // output row parity
  int pb = par & 1;                 // output col parity

  int lane = threadIdx.x & 31;
  int half = lane >> 4;             // K-pair selector
  int lidx = lane & 15;             // A row (out-ch) / B col (spatial)

  int o = ot * 16 + lidx;           // A-matrix row = out channel

  // Tap geometry (ty = half): a=0: ty0->(ky=1,row=i), ty1->(ky=3,row=i-1)
  //                           a=1: ty0->(ky=0,row=i+1), ty1->(ky=2,row=i)
  int kyL  = (half == 0) ? (a == 0 ? 1 : 0) : (a == 0 ? 3 : 2);
  int rowL = (half == 0) ? (a == 0 ? i : i + 1) : (a == 0 ? i - 1 : i);
  bool rowOK = (rowL >= 0) && (rowL < IH);
  int kx0 = (pb == 0) ? 1 : 0;      // tx=0 weight col
  int kx1 = (pb == 0) ? 3 : 2;      // tx=1 weight col

  // Per-N-tile input columns + in-range flags (lidx-dependent, c-invariant).
  int c0[4], c1[4];
  bool ok0[4], ok1[4];
#pragma unroll
  for (int nt = 0; nt < 4; ++nt) {
    int j = nt * 16 + lidx;
    c0[nt] = (pb == 0) ? j : j + 1;
    c1[nt] = (pb == 0) ? j - 1 : j;
    ok0[nt] = rowOK && (c0[nt] >= 0) && (c0[nt] < IW);
    ok1[nt] = rowOK && (c1[nt] >= 0) && (c1[nt] < IW);
  }

  v8f acc[4] = {};
  int xrow = rowOK ? (((bb * INCH) * IH + rowL) * IW) : 0;
  int wrow = (o << 4) + (kyL << 2);

  for (int c = 0; c < INCH; ++c) {
    int wbase = (c * INCH << 4) + wrow;     // Wd[c][o][kyL][*]
    int xbase = xrow + c * (IH * IW);       // X[bb][c][rowL][*]
    __builtin_prefetch((const void*)(Wd + wbase + (PFDIST * INCH << 4)), 0, 0);
    __builtin_prefetch((const void*)(X + xbase + PFDIST * (IH * IW)), 0, 0);
    v2f av;
    av.x = Wd[wbase + kx0];
    av.y = Wd[wbase + kx1];
#pragma unroll
    for (int nt = 0; nt < 4; ++nt) {
      v2f bv;
      bv.x = ok0[nt] ? X[xbase + c0[nt]] : 0.0f;
      bv.y = ok1[nt] ? X[xbase + c1[nt]] : 0.0f;
      acc[nt] = __builtin_amdgcn_wmma_f32_16x16x4_f32(false, av, false, bv,
                                                      (short)0, acc[nt],
                                                      false, false);
    }
  }

  // Store: C VGPR r -> row m = r + 8*half, col = lidx
  int y = 2 * i + a;
#pragma unroll
  for (int nt = 0; nt < 4; ++nt) {
    int xx = 2 * (nt * 16 + lidx) + pb;
#pragma unroll
    for (int r = 0; r < 8; ++r) {
      int oo = ot * 16 + r + 8 * half;
      Out[((bb * INCH + oo) * HH + y) * WW + xx] = acc[nt][r];
    }
  }
}

// ---------------------------------------------------------------------------
// InstanceNorm(eps=1e-5) + SiLU, in place, one block per (b, channel).
// ---------------------------------------------------------------------------
__global__ void __launch_bounds__(256) hh_instnorm_silu(float* __restrict__ d) {
  float* p = d + (size_t)blockIdx.x * SP;
  __shared__ float s1[256], s2[256];
  float sa = 0.0f, sb = 0.0f;
  for (int i = threadIdx.x; i < SP; i += 256) {
    float v = p[i];
    sa += v;
    sb += v * v;
  }
  s1[threadIdx.x] = sa;
  s2[threadIdx.x] = sb;
  __syncthreads();
  for (int off = 128; off > 0; off >>= 1) {
    if (threadIdx.x < off) {
      s1[threadIdx.x] += s1[threadIdx.x + off];
      s2[threadIdx.x] += s2[threadIdx.x + off];
    }
    __syncthreads();
  }
  float mean = s1[0] * (1.0f / SP);
  float var  = s2[0] * (1.0f / SP) - mean * mean;
  float inv  = rsqrtf(var + 1e-5f);
  for (int i = threadIdx.x; i < SP; i += 256) {
    float z = (p[i] - mean) * inv;
    p[i] = z / (1.0f + expf(-z));            // SiLU
  }
}

// ---------------------------------------------------------------------------
// 1x1 conv as GEMM: Out[b,o,s] = sum_k W[o,k]*H[b,k,s] + bias[o].
// Wave per 16x64 (M x N) tile: 4 accumulators reuse each A (weight) fetch.
// ---------------------------------------------------------------------------
__global__ void __launch_bounds__(256) hh_gemm1x1_wmma(
    const float* __restrict__ W, const float* __restrict__ bias,
    const float* __restrict__ Hin, float* __restrict__ Out,
    int M, int Mtiles) {
  int t = blockIdx.x * 8 + (threadIdx.x >> 5);
  int sg = t % 192;      t /= 192;  // 12288/64 spatial groups
  int mt = t % Mtiles;   t /= Mtiles;
  int bb = t;

  int lane = threadIdx.x & 31;
  int half = lane >> 4;
  int lidx = lane & 15;

  int o = mt * 16 + lidx;                 // A row (clamp addr for M-tail;
  const float* Wr = W + ((o < M) ? o : 0) * INCH;  // garbage rows never stored)
  int s0 = sg * 64 + lidx;                // first B col of this wave
  const float* Hb = Hin + (size_t)bb * INCH * SP + s0;

  v8f acc[4] = {};
  for (int k0 = 0; k0 < INCH; k0 += 4) {
    int kk = k0 + 2 * half;               // even -> 8B aligned pair
    __builtin_prefetch((const void*)(Wr + kk + 4 * PFDIST), 0, 0);
    __builtin_prefetch((const void*)(Hb + (size_t)(kk + 4 * PFDIST) * SP), 0, 0);
    v2f av;
    av.x = Wr[kk];
    av.y = Wr[kk + 1];
#pragma unroll
    for (int nt = 0; nt < 4; ++nt) {
      v2f bv;
      bv.x = Hb[(size_t)kk * SP + nt * 16];
      bv.y = Hb[(size_t)(kk + 1) * SP + nt * 16];
      acc[nt] = __builtin_amdgcn_wmma_f32_16x16x4_f32(false, av, false, bv,
                                                      (short)0, acc[nt],
                                                      false, false);
    }
  }

#pragma unroll
  for (int nt = 0; nt < 4; ++nt) {
#pragma unroll
    for (int r = 0; r < 8; ++r) {
      int oo = mt * 16 + r + 8 * half;
      if (oo < M)
        Out[(size_t)bb * M * SP + (size_t)oo * SP + s0 + nt * 16] =
            acc[nt][r] + bias[oo];
    }
  }
}

// ---------------------------------------------------------------------------
// Postprocess: one block per (b,k). argmax/score, Gaussian blur (sigma=2,
// radius=8, zero border=5, symmetric pad=8), rescale, log-clip, 3x3 Taylor.
// ---------------------------------------------------------------------------
#define PH 106            // 96 + 2*5
#define PW 138            // 128 + 2*5
#define PWP 140           // padded LDS row stride

__device__ __forceinline__ int sym_idx(int i, int n) {
  if (i < 0) return -1 - i;
  if (i >= n) return 2 * n - 1 - i;
  return i;
}

__global__ void __launch_bounds__(256) hh_postprocess(
    const float* __restrict__ hm, float* __restrict__ out) {
  extern __shared__ float tmp[];          // PH * PWP vertical-pass buffer
  __shared__ float sv[256];
  __shared__ int si[256];
  int bc = blockIdx.x;                    // b*133 + k
  const float* m = hm + (size_t)bc * SP;
  int tid = threadIdx.x;

  // Gaussian taps (radius 8, sigma 2), normalized.
  float g[17];
  {
    float gs = 0.0f;
    for (int t = 0; t < 17; ++t) {
      float d = (float)(t - 8) * 0.5f;    // t/sigma
      g[t] = expf(-0.5f * d * d);
      gs += g[t];
    }
    float r = 1.0f / gs;
    for (int t = 0; t < 17; ++t) g[t] *= r;
  }

  // --- argmax (first occurrence) + score ---
  float bv = -INFINITY;
  int bi = 0;
  for (int i = tid; i < SP; i += 256) {
    float v = m[i];
    if (v > bv) { bv = v; bi = i; }
  }
  sv[tid] = bv; si[tid] = bi;
  __syncthreads();
  for (int off = 128; off > 0; off >>= 1) {
    if (tid < off) {
      float ov = sv[tid + off]; int oi = si[tid + off];
      if (ov > sv[tid] || (ov == sv[tid] && oi < si[tid])) { sv[tid] = ov; si[tid] = oi; }
    }
    __syncthreads();
  }
  float score = sv[0];
  int bidx = si[0];
  __syncthreads();

  // --- vertical blur pass into LDS (zero-padded domain, symmetric edges) ---
  for (int p = tid; p < PH * PW; p += 256) {
    int yy = p / PW, xx = p % PW;
    float acc = 0.0f;
    if (xx >= 5 && xx < 133) {
      for (int t = 0; t < 17; ++t) {
        int y = sym_idx(yy + t - 8, PH);
        if (y >= 5 && y < 101)
          acc += g[t] * m[(y - 5) * WW + (xx - 5)];
      }
    }
    tmp[yy * PWP + xx] = acc;
  }
  __syncthreads();

  // --- horizontal pass fused with cmax reduction over the 96x128 crop ---
  float cm = -INFINITY;
  for (int p = tid; p < SP; p += 256) {
    int y = p >> 7, x = p & 127;
    int rb = (y + 5) * PWP;
    float acc = 0.0f;
    for (int t = 0; t < 17; ++t) {
      int xi = sym_idx(x + 5 + t - 8, PW);
      acc += g[t] * tmp[rb + xi];
    }
    cm = fmaxf(cm, acc);
  }
  sv[tid] = cm;
  __syncthreads();
  for (int off = 128; off > 0; off >>= 1) {
    if (tid < off) sv[tid] = fmaxf(sv[tid], sv[tid + off]);
    __syncthreads();
  }
  float cmax = sv[0];

  // --- Taylor refinement (single lane) ---
  if (tid == 0) {
    int ys = bidx >> 7, xs = bidx & 127;
    float scale = (cmax > 0.0f) ? (score / cmax) : 1.0f;
    float lg[3][3];
    for (int dy = -1; dy <= 1; ++dy) {
      for (int dx = -1; dx <= 1; ++dx) {
        int yy = min(max(ys + dy, 0), HH - 1);   // edge pad
        int xc = min(max(xs + dx, 0), WW - 1);
        int rb = (yy + 5) * PWP;
        float v = 0.0f;
        for (int t = 0; t < 17; ++t) {
          int xi = sym_idx(xc + 5 + t - 8, PW);
          v += g[t] * tmp[rb + xi];
        }
        if (cmax > 0.0f) v *= scale;
        v = fminf(fmaxf(v, 0.001f), 50.0f);
        lg[dy + 1][dx + 1] = logf(v);
      }
    }
    float dxv = 0.5f * (lg[1][2] - lg[1][0]);
    float dyv = 0.5f * (lg[2][1] - lg[0][1]);
    float dxx = lg[1][2] - 2.0f * lg[1][1] + lg[1][0];
    float dyy = lg[2][1] - 2.0f * lg[1][1] + lg[0][1];
    float dxy = 0.5f * (lg[2][2] - lg[1][2] - lg[2][1] + 2.0f * lg[1][1] -
                        lg[1][0] - lg[0][1] + lg[0][0]);
    float aa = dxx + EPS_F32;
    float dd = dyy + EPS_F32;
    float b2 = dxy;
    float det = aa * dd - b2 * b2;
    float ox = (dd * dxv - b2 * dyv) / det;
    float oy = (-b2 * dxv + aa * dyv) / det;
    float kx = ((float)xs - ox) * (383.0f / 95.0f);   // SCALE_FACTOR[0]
    float ky = ((float)ys - oy) * (511.0f / 127.0f);  // SCALE_FACTOR[1]
    if (score <= 0.0f) { kx = -1.0f; ky = -1.0f; }
    out[bc * 2 + 0] = kx;
    out[bc * 2 + 1] = ky;
    out[2 * OUTCH * 2 + bc] = score;                  // scores after kp block
  }
}

// ---------------------------------------------------------------------------
extern "C" void kernel_launch(void* const* d_in, const int* in_sizes, int n_in,
                              void* d_out, int out_size, void* d_ws, size_t ws_size,
                              hipStream_t stream) {
  const float* X  = (const float*)d_in[0];   // [2,640,48,64]
  const float* Wd = (const float*)d_in[1];   // [640,640,4,4] (in,out,kh,kw)
  const float* Wc = (const float*)d_in[2];   // [640,640,1,1] (o,c)
  const float* Bc = (const float*)d_in[3];   // [640]
  const float* Wf = (const float*)d_in[4];   // [133,640,1,1]
  const float* Bf = (const float*)d_in[5];   // [133]
  float* ws = (float*)d_ws;

  float* h1 = ws;
  float* h2 = ws + (size_t)2 * INCH * SP;    // +15,728,640 floats
  float* hm = ws + (size_t)4 * INCH * SP;    // +31,457,280 floats

  // Stage A: deconv (4 par x 2 batch x 40 o-tiles x 48 rows) / 8 waves
  hh_deconv_wmma<<<1920, 256, 0, stream>>>(X, Wd, h1);
  // Stage B: IN + SiLU in place
  hh_instnorm_silu<<<2 * INCH, 256, 0, stream>>>(h1);
  // Stage C: 1x1 conv 640->640 (2 x 40 x 192) / 8 waves
  hh_gemm1x1_wmma<<<1920, 256, 0, stream>>>(Wc, Bc, h1, h2, INCH, 40);
  // Stage D: IN + SiLU in place
  hh_instnorm_silu<<<2 * INCH, 256, 0, stream>>>(h2);
  // Stage E: final 1x1 conv 640->133 (2 x 9 x 192) / 8 waves, tail-guarded
  hh_gemm1x1_wmma<<<432, 256, 0, stream>>>(Wf, Bf, h2, hm, OUTCH, 9);
  // Stage F: argmax + blur + Taylor refinement -> kp, scores
  hh_postprocess<<<2 * OUTCH, 256, PH * PWP * sizeof(float), stream>>>(
      hm, (float*)d_out);
}